// NoShareConv2d_91328184582801
// MI455X (gfx1250) — compile-verified
//
#include <hip/hip_runtime.h>

typedef float v2f __attribute__((ext_vector_type(2)));
typedef float v8f __attribute__((ext_vector_type(8)));

#define BATCH 32
#define CIN   64
#define HW    32
#define OUT   64
#define INDIM 576          // CIN * 3 * 3
#define NPOS  1024         // L*L
#define STR   580          // padded LDS row stride in floats (580 % 64 == 4 -> conflict-free)

__global__ __launch_bounds__(256, 1)
void noshare_conv2d_wmma_f32(const float* __restrict__ x,
                             const float* __restrict__ weight,
                             const float* __restrict__ bias,
                             float* __restrict__ out)
{
    extern __shared__ float sA[];            // [BATCH][STR] im2col panel
    const int p   = blockIdx.x;              // spatial position 0..1023
    const int pi  = p >> 5;
    const int pj  = p & 31;
    const int tid = threadIdx.x;

    // ---------- Phase 1: cooperative im2col (32 x 576) into LDS ----------
    // 18432 elements / 256 threads = 72 full iterations (no divergence).
    for (int e = tid; e < BATCH * INDIM; e += 256) {
        int b  = e / INDIM;
        int i  = e - b * INDIM;              // channel-major patch index (torch Unfold order)
        int c  = i / 9;
        int r  = i - c * 9;
        int ki = r / 3;
        int kj = r - ki * 3;
        int y  = pi + ki - 1;                // padding = 1
        int xx = pj + kj - 1;
        float v = 0.0f;
        if ((unsigned)y < 32u && (unsigned)xx < 32u)
            v = x[(((size_t)b * CIN + c) * HW + y) * HW + xx];
        sA[b * STR + i] = v;
    }
    __syncthreads();

    // ---------- Phase 2: one 16x16 C-tile per wave, K=576 via 16x16x4 f32 WMMA ----------
    const int lane = tid & 31;
    const int wave = tid >> 5;               // 0..7
    const int mt   = wave >> 2;              // batch tile   (0..1)
    const int nt   = wave & 3;               // channel tile (0..3)
    const int h    = lane >> 4;              // lane half: selects K {0,1} vs {2,3}
    const int ln   = lane & 15;

    // A fragment source: cols[M][k0 + 2h + {0,1}] -> one ds_load_b64 per lane per step
    const int    arow = mt * 16 + ln;
    const float* aptr = &sA[arow * STR + 2 * h];

    // B fragment source: w[k0 + 2h + {0,1}][n] -> two coalesced dword loads per lane
    const int    n    = nt * 16 + ln;
    const float* wptr = weight + (size_t)p * (INDIM * OUT) + (size_t)h * (2 * OUT) + n;

    v8f acc = {};
    #pragma unroll 4
    for (int k0 = 0; k0 < INDIM; k0 += 4) {
        v2f a = *(const v2f*)aptr;           // A: 16x4 f32 fragment (2 VGPRs)
        v2f b;
        b.x = wptr[0];                       // B row k0+2h
        b.y = wptr[OUT];                     // B row k0+2h+1
        // D = A*B + C  (8 args: neg_a, A, neg_b, B, c_mod, C, reuse_a, reuse_b)
        acc = __builtin_amdgcn_wmma_f32_16x16x4_f32(false, a, false, b,
                                                    (short)0, acc, false, false);
        aptr += 4;
        wptr += 4 * OUT;
    }

    // ---------- Phase 3: bias + store (C layout: VGPR v -> M = v + 8h, N = ln) ----------
    const float bv = bias[p * OUT + n];
    #pragma unroll
    for (int v = 0; v < 8; ++v) {
        int brow = mt * 16 + v + 8 * h;      // global batch index
        out[((size_t)brow * OUT + n) * NPOS + p] = acc[v] + bv;
    }
}

extern "C" void kernel_launch(void* const* d_in, const int* in_sizes, int n_in,
                              void* d_out, int out_size, void* d_ws, size_t ws_size,
                              hipStream_t stream) {
    const float* x = (const float*)d_in[0];   // [32, 64, 32, 32]
    const float* w = (const float*)d_in[1];   // [1024, 576, 64]
    const float* b = (const float*)d_in[2];   // [1024, 64]
    float* out = (float*)d_out;               // [32, 64, 32, 32]

    const size_t shmem = (size_t)BATCH * STR * sizeof(float);   // 74,240 B (< 320 KB WGP LDS)
    noshare_conv2d_wmma_f32<<<dim3(NPOS), dim3(256), shmem, stream>>>(x, w, b, out);
}